// FFA_5772436045850
// MI455X (gfx1250) — compile-verified
//
#include <hip/hip_runtime.h>

// ---------------------------------------------------------------------------
// FFA forward, CDNA5 (gfx1250).
//
// Math:  out[t] = g * out[t-1] + x[t],  g[f,d] = exp(a_f + i*b_d),
//        out[-1] = memory[b,f,d].   (Equivalent to the reference's
//        gamma^outer * cumsum(gamma^inner * x) + memory * gamma^offset.)
//
// Roofline: output = 268 MB complex64, streamed once -> ~11.5 us floor at
// 23.3 TB/s HBM. Compute is ~0.3 GFLOP (irrelevant). The design goal is
// (a) break the T=1024 sequential chain (chunked parallel scan, WMMA-computed
// chunk prefixes) and (b) keep the store path streaming (non-temporal hints,
// x staged to LDS via the async engine so the recurrence never waits on
// global latency).
//
// Per block = one (b, 16-f, 16-d) tile over all T:
//   Stage A: 8 waves compute 8 time-chunk sums S_c[f,d] = sum y[t,f]*w[t,d]
//            with V_WMMA_F32_16X16X4_F32 (real & imag accumulators), where
//            y = exp(a*inner_t)*x (real) and w = exp(i*b*inner_t) (unit
//            complex) are maintained by multiplicative recurrences.
//   Stage B: chunk prefix is a plain add in this basis (folded into C setup).
//   Stage C: each wave seeds h0 = gamma^outer*P + mem*gamma^offset and runs
//            the native recurrence h <- g*h + x_t for its 128 steps.
//            x is staged with GLOBAL_LOAD_ASYNC_TO_LDS_B128 (ASYNCcnt);
//            output written with non-temporal global_store_b64.
// ---------------------------------------------------------------------------

typedef float v2f __attribute__((ext_vector_type(2)));
typedef float v8f __attribute__((ext_vector_type(8)));

#define BB      2
#define TT      1024
#define FF      256
#define DD      64
#define MAXLEN  2048
#define LIMITF  0.03332170f     // log(float32 max)/2048 - 0.01
#define NW      8               // waves per block (wave32)
#define CH      (TT / NW)       // 128 timesteps per wave-chunk
#define HALF    64              // sub-chunk staged in LDS at a time

__device__ __forceinline__ float clamp_a(float v) {
    return fminf(fmaxf(v, -LIMITF), -1e-8f);
}

__global__ __launch_bounds__(256) void ffa_scan_kernel(
    const float* __restrict__ x,     // [B,T,F]
    const float* __restrict__ mre,   // [B,1,F,D]
    const float* __restrict__ mim,   // [B,1,F,D]
    const float* __restrict__ a,     // [F]
    const float* __restrict__ b,     // [D]
    float*       __restrict__ out)   // [B,T,F,D] complex64 as float pairs
{
    __shared__ float Ssh[NW][16][16][2];      // chunk sums, 16 KB
    __shared__ float xsh[NW][HALF][16];       // per-wave x staging, 32 KB

    const int tid  = threadIdx.x;
    const int wv   = tid >> 5;
    const int lane = tid & 31;

    const int FT = FF / 16, DT = DD / 16;
    const int bid = blockIdx.x;
    const int bb  = bid / (FT * DT);
    const int ft  = (bid / DT) % FT;
    const int dt  = bid % DT;
    const int f0  = ft * 16, d0 = dt * 16;

    const int t0 = wv * CH;                   // this wave's time chunk

    // =======================================================================
    // Stage A: S_wv[f,d] = sum_{t in chunk} y[t,f] * w[t,d] via WMMA f32 K=4.
    // A-frag (16x4): lane m = lane&15 -> f row, khalf = lane>>4 -> k in {2kh,2kh+1}
    // B-frag (4x16): lane n = lane&15 -> d col, same k slots per VGPR pair.
    // =======================================================================
    const int mrow  = lane & 15;
    const int khalf = lane >> 4;

    const float afA = clamp_a(a[f0 + mrow]);
    const float bdB = b[d0 + mrow];

    // y-scale recurrence: eys = exp(afA * inner(t)), inner(t) = 2047 - t.
    const float inn0 = (float)(MAXLEN - 1 - (t0 + 2 * khalf));
    float eys = expf(afA * inn0);
    const float ema1 = expf(-afA);                 // ratio per +1 step in t
    const float ema4 = (ema1 * ema1) * (ema1 * ema1);

    // w recurrence: c = exp(i * bdB * inner(t)); u1 = e^{-i*bd}, u4 = e^{-4i*bd}
    float s0, c0;  sincosf(bdB * inn0, &s0, &c0);
    float cre = c0, cim = s0;
    float sb, cb;  sincosf(bdB, &sb, &cb);
    const float u1re = cb,  u1im = -sb;
    float s4, c4;  sincosf(4.0f * bdB, &s4, &c4);
    const float u4re = c4,  u4im = -s4;

    v8f accRe = {}, accIm = {};
    const float* xpA = x + ((size_t)bb * TT + (t0 + 2 * khalf)) * FF + f0 + mrow;

    #pragma unroll 4
    for (int i = 0; i < CH / 4; ++i) {
        __builtin_prefetch(xpA + 8 * FF, 0, 0);            // global_prefetch_b8
        const float xa0 = xpA[0];
        const float xa1 = xpA[FF];
        xpA += 4 * FF;

        v2f Af  = { eys * xa0, (eys * ema1) * xa1 };
        const float c1re = cre * u1re - cim * u1im;
        const float c1im = cre * u1im + cim * u1re;
        v2f Bre = { cre, c1re };
        v2f Bim = { cim, c1im };

        accRe = __builtin_amdgcn_wmma_f32_16x16x4_f32(
                    false, Af, false, Bre, (short)0, accRe, false, false);
        accIm = __builtin_amdgcn_wmma_f32_16x16x4_f32(
                    false, Af, false, Bim, (short)0, accIm, false, false);

        // advance 4 timesteps
        eys *= ema4;
        const float nre = cre * u4re - cim * u4im;
        const float nim = cre * u4im + cim * u4re;
        cre = nre; cim = nim;
    }

    // Spill C-layout accumulators to LDS at absolute (f,d) tile coordinates:
    // VGPR r: lanes 0-15 hold M=r, lanes 16-31 hold M=r+8; N = lane&15.
    #pragma unroll
    for (int r = 0; r < 8; ++r) {
        const int m = khalf ? (r + 8) : r;
        Ssh[wv][m][mrow][0] = accRe[r];
        Ssh[wv][m][mrow][1] = accIm[r];
    }
    __syncthreads();

    // =======================================================================
    // Stage C: seed chunk state and run native recurrence h <- g*h + x_t.
    // Lane owns d = d0 + (lane&15), and 8 f's: f = f0 + (lane>>4)*8 + j.
    // =======================================================================
    const int dcol = lane & 15;
    const int fgrp = (lane >> 4) * 8;
    const float bd = b[d0 + dcol];

    float sgb, cgb; sincosf(bd, &sgb, &cgb);          // e^{i*bd}
    float so, co;   sincosf(bd * (float)(t0 - MAXLEN), &so, &co);  // U_out
    float sf, cf;   sincosf(bd * (float)(t0),          &sf, &cf);  // U_off

    float hre[8], him[8], gre[8], gim[8];
    #pragma unroll
    for (int j = 0; j < 8; ++j) {
        const int fi = fgrp + j;
        const int f  = f0 + fi;
        const float af = clamp_a(a[f]);
        const float eg = expf(af);
        gre[j] = eg * cgb;
        gim[j] = eg * sgb;

        // prefix over earlier chunks (plain add in the y/w basis)
        float Pre = 0.f, Pim = 0.f;
        for (int c = 0; c < wv; ++c) {
            Pre += Ssh[c][fi][dcol][0];
            Pim += Ssh[c][fi][dcol][1];
        }
        const float Eo = expf(af * (float)(t0 - MAXLEN));   // gamma^outer(t0-1) mag
        const float Ef = expf(af * (float)(t0));            // gamma^offset(t0-1) mag

        const size_t mix = ((size_t)bb * FF + f) * DD + (d0 + dcol);
        const float mr = mre[mix], mi = mim[mix];

        // h(-1 of chunk) = Eo*(Uo ox P) + Ef*(Uf ox mem)
        hre[j] = Eo * (co * Pre - so * Pim) + Ef * (cf * mr - sf * mi);
        him[j] = Eo * (co * Pim + so * Pre) + Ef * (cf * mi + sf * mr);
    }

    v2f* __restrict__ outv = (v2f*)out;

    for (int half = 0; half < 2; ++half) {
        const int tb = t0 + half * HALF;

        // Make sure prior ds_loads on this region drained before the async
        // engine overwrites it (DS is in-order per wave, but unordered vs the
        // async unit).
        asm volatile("s_wait_dscnt 0x0" ::: "memory");

        // Stage this wave's x sub-block [HALF][16] into LDS through the
        // async path: GLOBAL_LOAD_ASYNC_TO_LDS_B128 (memory -> LDS, no VGPR
        // round-trip, tracked by ASYNCcnt). 8 x 32 lanes x 16B = 4 KB.
        #pragma unroll
        for (int v = 0; v < 8; ++v) {
            const int flat = lane + 32 * v;           // 0..255 float4 slots
            const int tl = flat >> 2, q4 = (flat & 3) * 4;
            const unsigned long long gaddr = (unsigned long long)(uintptr_t)
                (x + ((size_t)bb * TT + tb + tl) * FF + f0 + q4);
            const unsigned int laddr =
                (unsigned int)(uintptr_t)&xsh[wv][tl][q4];
            asm volatile("global_load_async_to_lds_b128 %0, %1, off"
                         :: "v"(laddr), "v"(gaddr) : "memory");
        }
        asm volatile("s_wait_asynccnt 0x0" ::: "memory");

        for (int tl = 0; tl < HALF; ++tl) {
            const int t = tb + tl;
            const float4 xlo = *(const float4*)&xsh[wv][tl][fgrp];
            const float4 xhi = *(const float4*)&xsh[wv][tl][fgrp + 4];
            const float xv[8] = { xlo.x, xlo.y, xlo.z, xlo.w,
                                  xhi.x, xhi.y, xhi.z, xhi.w };
            const size_t orow = ((size_t)bb * TT + t) * FF;
            #pragma unroll
            for (int j = 0; j < 8; ++j) {
                // h <- g*h + x  (complex FMA; x is real)
                const float tr = fmaf(gre[j], hre[j], xv[j]);
                const float nr = fmaf(-gim[j], him[j], tr);
                const float ni = fmaf(gre[j], him[j], gim[j] * hre[j]);
                hre[j] = nr; him[j] = ni;
                // Streaming store: 268 MB written once, never re-read ->
                // non-temporal hint keeps it from displacing L2.
                v2f o = { nr, ni };
                __builtin_nontemporal_store(
                    o, &outv[(orow + (f0 + fgrp + j)) * DD + (d0 + dcol)]);
            }
        }
    }
}

extern "C" void kernel_launch(void* const* d_in, const int* in_sizes, int n_in,
                              void* d_out, int out_size, void* d_ws, size_t ws_size,
                              hipStream_t stream) {
    const float* x   = (const float*)d_in[0];
    const float* mre = (const float*)d_in[1];
    const float* mim = (const float*)d_in[2];
    const float* a   = (const float*)d_in[3];
    const float* b   = (const float*)d_in[4];

    const int grid = BB * (FF / 16) * (DD / 16);   // 128 blocks
    ffa_scan_kernel<<<grid, 256, 0, stream>>>(x, mre, mim, a, b, (float*)d_out);
}